// mHC_84250078478524
// MI455X (gfx1250) — compile-verified
//
#include <hip/hip_runtime.h>
#include <hip/hip_bf16.h>
#include <math.h>

// Problem constants (from reference): M=8192, K=8192 (DIM=2048 * N_STREAM=4), N=32.
#define M_TOT 8192
#define K_TOT 8192
#define DIM_D 2048
#define WAVES 8           // 256-thread block = 8 wave32
#define KSLICE (K_TOT / WAVES)

typedef __attribute__((ext_vector_type(2))) float v2f;
typedef __attribute__((ext_vector_type(8))) float v8f;

// ---------------------------------------------------------------------------
// Pass 1: one block = 16 rows of x. Each wave covers a 1024-wide K slice,
// computing a partial 16x32 H tile with V_WMMA_F32_16X16X4_F32 and a partial
// sum-of-squares. LDS reduction (fixed order) -> per-row sigmoid/Sinkhorn
// math -> fused 4x4 mixing matrix W[m] written to workspace.
// ---------------------------------------------------------------------------
__global__ __launch_bounds__(256) void mhc_pass1(const float* __restrict__ x,
                                                 const float* __restrict__ phi,
                                                 const float* __restrict__ ab,
                                                 float* __restrict__ Wg) {
    __shared__ float lds_H[WAVES][16][32];   // per-wave partial H tiles
    __shared__ float lds_ssq[WAVES][32];     // per-lane partial sum(x^2)
    __shared__ float Hs[16][32];             // reduced H tile
    __shared__ float ssq[16];                // reduced row sumsq

    const int tid   = threadIdx.x;
    const int wave  = tid >> 5;
    const int lane  = tid & 31;
    const int row   = lane & 15;             // M index within tile (0..15)
    const int khalf = lane >> 4;             // 0: K={0,1}, 1: K={2,3}
    const int m0    = blockIdx.x * 16;

    v8f acc0 = {};                           // N = 0..15
    v8f acc1 = {};                           // N = 16..31
    float ss = 0.0f;

    const float* xrow = x + (size_t)(m0 + row) * K_TOT;
    const int kBegin = wave * KSLICE;
    const int kEnd   = kBegin + KSLICE;

    for (int k = kBegin; k < kEnd; k += 4) {
        // A tile 16x4 f32: lane (row,khalf) holds x[m0+row, k+2*khalf + {0,1}]
        v2f a = *(const v2f*)(xrow + k + 2 * khalf);
        ss += a.x * a.x + a.y * a.y;         // every x element counted once

        // B tiles 4x16 f32 (phi row-major, stride 32):
        // v.x = phi[k+2*khalf,   n0+row], v.y = phi[k+2*khalf+1, n0+row]
        const float* prow = phi + (size_t)(k + 2 * khalf) * 32;
        v2f b0, b1;
        b0.x = prow[row];
        b0.y = prow[32 + row];
        b1.x = prow[16 + row];
        b1.y = prow[32 + 16 + row];

        acc0 = __builtin_amdgcn_wmma_f32_16x16x4_f32(
            false, a, false, b0, (short)0, acc0, false, false);
        acc1 = __builtin_amdgcn_wmma_f32_16x16x4_f32(
            false, a, false, b1, (short)0, acc1, false, false);
    }

    // Spill per-wave accumulators to LDS using the documented C/D layout:
    // VGPR e, lanes 0-15 -> M=e, N=lane; lanes 16-31 -> M=e+8, N=lane-16.
#pragma unroll
    for (int e = 0; e < 8; ++e) {
        lds_H[wave][e + 8 * khalf][row]      = acc0[e];
        lds_H[wave][e + 8 * khalf][16 + row] = acc1[e];
    }
    lds_ssq[wave][lane] = ss;
    __syncthreads();

    // Fixed-order cross-wave reduction (deterministic).
    for (int idx = tid; idx < 16 * 32; idx += 256) {
        int r = idx >> 5, c = idx & 31;
        float s = 0.0f;
#pragma unroll
        for (int w = 0; w < WAVES; ++w) s += lds_H[w][r][c];
        Hs[r][c] = s;
    }
    if (tid < 16) {
        float s = 0.0f;
#pragma unroll
        for (int w = 0; w < WAVES; ++w) s += lds_ssq[w][tid] + lds_ssq[w][tid + 16];
        ssq[tid] = s;
    }
    __syncthreads();

    // Per-row scalar math: one thread per row.
    if (tid < 16) {
        const int r = tid;
        const float inv_sqrtK = 1.0f / sqrtf((float)K_TOT);
        float rms = sqrtf(ssq[r]) * inv_sqrtK + 1e-9f;
        float inv = 1.0f / rms;

        float a_res = ab[24], a_pre = ab[25], a_pos = ab[26];

        float spre[4], spos[4];
#pragma unroll
        for (int j = 0; j < 4; ++j) {
            float z1 = inv * a_pre * Hs[r][16 + j] + ab[16 + j];
            spre[j] = 1.0f / (1.0f + __expf(-z1));
            float z2 = inv * a_pos * Hs[r][20 + j] + ab[20 + j];
            spos[j] = 2.0f / (1.0f + __expf(-z2));
        }

        // Sinkhorn-Knopp, 1 iteration: exp -> row-normalize -> col-normalize.
        float S[4][4];
#pragma unroll
        for (int i = 0; i < 4; ++i) {
#pragma unroll
            for (int j = 0; j < 4; ++j)
                S[i][j] = __expf(inv * a_res * Hs[r][i * 4 + j] + ab[i * 4 + j]);
        }
#pragma unroll
        for (int i = 0; i < 4; ++i) {
            float rs = S[i][0] + S[i][1] + S[i][2] + S[i][3] + 1e-20f;
            float rr = 1.0f / rs;
#pragma unroll
            for (int j = 0; j < 4; ++j) S[i][j] *= rr;
        }
#pragma unroll
        for (int j = 0; j < 4; ++j) {
            float cs = S[0][j] + S[1][j] + S[2][j] + S[3][j] + 1e-20f;
            float cr = 1.0f / cs;
#pragma unroll
            for (int i = 0; i < 4; ++i) S[i][j] *= cr;
        }

        // Fused mixing matrix: W = S + outer(spos, spre).
        float* wout = Wg + (size_t)(m0 + r) * 16;
#pragma unroll
        for (int i = 0; i < 4; ++i)
#pragma unroll
            for (int j = 0; j < 4; ++j)
                wout[i * 4 + j] = S[i][j] + spos[i] * spre[j];
    }
}

// ---------------------------------------------------------------------------
// Pass 2: out[m, i*D + d] = sum_j W[m,i,j] * x[m, j*D + d].
// Pure streaming pass: float4 (b128) loads/stores, 64B in + 64B out per thread.
// grid = (D / (256*4), M); one block handles 1024 contiguous d's of one row.
// ---------------------------------------------------------------------------
__global__ __launch_bounds__(256) void mhc_pass2(const float* __restrict__ x,
                                                 const float* __restrict__ Wg,
                                                 float* __restrict__ out) {
    __shared__ float Wl[16];
    const int m   = blockIdx.y;
    const int tid = threadIdx.x;
    if (tid < 16) Wl[tid] = Wg[(size_t)m * 16 + tid];
    __syncthreads();

    const int d = (blockIdx.x * 256 + tid) * 4;     // 0..D-4
    const size_t base = (size_t)m * K_TOT;

    float4 xv[4];
#pragma unroll
    for (int j = 0; j < 4; ++j)
        xv[j] = *(const float4*)(x + base + (size_t)j * DIM_D + d);

#pragma unroll
    for (int i = 0; i < 4; ++i) {
        float w0 = Wl[i * 4 + 0], w1 = Wl[i * 4 + 1];
        float w2 = Wl[i * 4 + 2], w3 = Wl[i * 4 + 3];
        float4 o;
        o.x = w0 * xv[0].x + w1 * xv[1].x + w2 * xv[2].x + w3 * xv[3].x;
        o.y = w0 * xv[0].y + w1 * xv[1].y + w2 * xv[2].y + w3 * xv[3].y;
        o.z = w0 * xv[0].z + w1 * xv[1].z + w2 * xv[2].z + w3 * xv[3].z;
        o.w = w0 * xv[0].w + w1 * xv[1].w + w2 * xv[2].w + w3 * xv[3].w;
        *(float4*)(out + base + (size_t)i * DIM_D + d) = o;
    }
}

extern "C" void kernel_launch(void* const* d_in, const int* in_sizes, int n_in,
                              void* d_out, int out_size, void* d_ws, size_t ws_size,
                              hipStream_t stream) {
    (void)in_sizes; (void)n_in; (void)out_size; (void)ws_size;
    const float* x   = (const float*)d_in[0];   // (8192, 8192) fp32
    const float* phi = (const float*)d_in[1];   // (8192, 32) fp32
    const float* ab  = (const float*)d_in[2];   // (32,) fp32
    float* out = (float*)d_out;                 // (8192, 8192) fp32
    float* W   = (float*)d_ws;                  // needs 8192*16*4 = 512 KB scratch

    // Pass 1: 512 blocks x 256 threads (8 waves), one 16-row tile per block.
    mhc_pass1<<<dim3(M_TOT / 16), 256, 0, stream>>>(x, phi, ab, W);
    // Pass 2: (2, 8192) blocks x 256 threads; float4-vectorized streaming mix.
    mhc_pass2<<<dim3(DIM_D / (256 * 4), M_TOT), 256, 0, stream>>>(x, W, out);
}